// DeformableAttention_7541962572416
// MI455X (gfx1250) — compile-verified
//
#include <hip/hip_runtime.h>
#include <hip/hip_bf16.h>

typedef float v2f __attribute__((ext_vector_type(2)));
typedef float v8f __attribute__((ext_vector_type(8)));
typedef int   v4i __attribute__((ext_vector_type(4)));

typedef __attribute__((address_space(1))) v4i* gv4i_ptr;   // global int4*
typedef __attribute__((address_space(3))) v4i* lv4i_ptr;   // LDS int4*

#define D_MODEL   256
#define N_HEADS   8
#define N_LEVELS  4
#define N_POINTS  4
#define HEAD_DIM  32
#define LEN_IN    21760
#define LEN_Q     21760
#define BATCH     2
#define M_TOTAL   (BATCH * LEN_Q)   // 43520, divisible by 16

// CDNA5 async global->LDS staging (ASYNCcnt path). Guarded so the file still
// compiles if this toolchain doesn't declare the builtin.
#if defined(__gfx1250__) && __has_builtin(__builtin_amdgcn_global_load_async_to_lds_b128)
#define HAVE_ASYNC_LDS 1
#else
#define HAVE_ASYNC_LDS 0
#endif

#if defined(__gfx1250__) && __has_builtin(__builtin_amdgcn_s_wait_asynccnt)
#define WAIT_ASYNCCNT0() __builtin_amdgcn_s_wait_asynccnt(0)
#else
#define WAIT_ASYNCCNT0() asm volatile("s_wait_asynccnt 0x0" ::: "memory")
#endif

// ---------------------------------------------------------------------------
// fp32 WMMA GEMM:  D[M,N] = A[M,K] * B[K,N] + bias[N], optional row zero-mask.
// One wave -> one 16x16 D tile via V_WMMA_F32_16X16X4_F32 (K/4 steps).
// 8 waves/block share one tile_m, so the 16xK A slab (contiguous 16 KB for
// K=256) is staged ONCE per block into LDS with async global->LDS DMA; the
// per-step A operand then comes from ds_load_b64 instead of 8x-redundant
// global loads. B (<=256 KB weight matrix) stays L2-resident.
//
// f32 16x4 A layout : lane%16 -> M, half-wave picks K pair (VGPR0/1 = k, k+1)
// f32 4x16 B layout : lane%16 -> N, half-wave picks K pair
// f32 16x16 C/D     : lane%16 -> N, VGPR r -> M = r + 8*(lane/16)
// ---------------------------------------------------------------------------
template<int N, int K, bool MASKED>
__global__ __launch_bounds__(256)
void wmma_gemm_f32(const float* __restrict__ A,
                   const float* __restrict__ B,
                   const float* __restrict__ bias,
                   float* __restrict__ D,
                   const unsigned char* __restrict__ row_mask)
{
    __shared__ float Als[16 * K];   // 16 KB for K=256

    const int t      = threadIdx.x;
    const int lane   = t & 31;
    const int wave   = t >> 5;
    const int tile_n = blockIdx.x * 8 + wave;   // grids sized exactly; no guard
    const int tile_m = blockIdx.y;

    // ---- cooperative A-slab stage: rows [tile_m*16, +16) x [0,K) ----
    {
        const float* gsrc = A + (size_t)tile_m * 16 * K;
#if HAVE_ASYNC_LDS
        gv4i_ptr gp = (gv4i_ptr)gsrc;       // C-style cast: drops const, -> AS1
        lv4i_ptr lp = (lv4i_ptr)Als;        // -> AS3
#pragma unroll
        for (int i = 0; i < (16 * K) / (256 * 4); ++i) {   // 4 iters @ K=256
            const int e = t + i * 256;                     // 16B elements
            __builtin_amdgcn_global_load_async_to_lds_b128(gp + e, lp + e, 0, 0);
        }
        WAIT_ASYNCCNT0();
#else
#pragma unroll
        for (int i = 0; i < (16 * K) / (256 * 4); ++i) {
            const int e = (t + i * 256) * 4;
            *(float4*)&Als[e] = *(const float4*)&gsrc[e];
        }
#endif
        __syncthreads();
    }

    const int nl = lane & 15;          // N (and A's M) sub-index
    const int kh = (lane >> 4) * 2;    // which K pair this half-wave owns

    const float* a_lds = &Als[nl * K + kh];
    const float* b_ptr = B + (size_t)kh * N + tile_n * 16 + nl;

    v8f c = {};
#pragma unroll 16
    for (int k = 0; k < K; k += 4) {
        v2f a = *(const v2f*)(a_lds + k);          // ds_load_b64
        v2f b;
        b.x = b_ptr[(size_t)k * N];                // immediate-offset global loads
        b.y = b_ptr[(size_t)(k + 1) * N];
        // 8 args: (neg_a, A, neg_b, B, c_mod, C, reuse_a, reuse_b)
        c = __builtin_amdgcn_wmma_f32_16x16x4_f32(false, a, false, b,
                                                  (short)0, c, false, false);
    }

    const float bval = bias[tile_n * 16 + nl];
    const int   mh   = (lane >> 4) * 8;
#pragma unroll
    for (int r = 0; r < 8; ++r) {
        const int m = tile_m * 16 + mh + r;
        float v = c[r] + bval;
        if (MASKED && row_mask[m]) v = 0.0f;
        D[(size_t)m * N + tile_n * 16 + nl] = v;
    }
}

// ---------------------------------------------------------------------------
// Deformable sampling core. One wave per (batch, query, head); lane = head dim.
// Each bilinear corner fetch is one coalesced 128B line from `value` (44.6 MB,
// resident in MI455X's 192 MB L2). Softmax over the 16 (level,point) logits is
// computed redundantly per lane from broadcast loads.
// ---------------------------------------------------------------------------
__global__ __launch_bounds__(256)
void msdeform_core(const float* __restrict__ value,    // [B, LEN_IN, H, hd]
                   const float* __restrict__ ref,      // [B, Lq, L, 2]
                   const float* __restrict__ off_raw,  // [B, Lq, H, L, P, 2]
                   const float* __restrict__ aw_raw,   // [B, Lq, H, L*P]
                   float* __restrict__ out)            // [B, Lq, H*hd]
{
    const int Wl_tab[4]    = {128, 64, 32, 16};
    const int Hl_tab[4]    = {128, 64, 32, 16};
    const int start_tab[4] = {0, 16384, 20480, 21504};

    const int lane = threadIdx.x & 31;
    const int gid  = blockIdx.x * 8 + (threadIdx.x >> 5);  // (b*Lq + q)*8 + h
    const int h    = gid & 7;
    const int bq   = gid >> 3;                              // b*Lq + q
    const int b    = bq / LEN_Q;

    // --- softmax over 16 logits ---
    const float* awp = aw_raw + (size_t)gid * 16;
    float w16[16];
    float mx = -1e30f;
#pragma unroll
    for (int i = 0; i < 16; ++i) { w16[i] = awp[i]; mx = fmaxf(mx, w16[i]); }
    float s = 0.0f;
#pragma unroll
    for (int i = 0; i < 16; ++i) { w16[i] = __expf(w16[i] - mx); s += w16[i]; }
    const float inv_s = 1.0f / s;

    const float* offp = off_raw + (size_t)gid * (N_LEVELS * N_POINTS * 2);
    const float* refp = ref + (size_t)bq * (N_LEVELS * 2);

    float acc = 0.0f;
#pragma unroll
    for (int l = 0; l < N_LEVELS; ++l) {
        const int   Wl = Wl_tab[l], Hl = Hl_tab[l];
        const float fW = (float)Wl, fH = (float)Hl;
        const float rx = refp[l * 2 + 0];
        const float ry = refp[l * 2 + 1];
        const float* vbase =
            value + ((size_t)(b * LEN_IN + start_tab[l]) * N_HEADS + h) * HEAD_DIM + lane;

#pragma unroll
        for (int p = 0; p < N_POINTS; ++p) {
            const float ox = offp[(l * N_POINTS + p) * 2 + 0];
            const float oy = offp[(l * N_POINTS + p) * 2 + 1];
            const float x  = (rx + ox / fW) * fW - 0.5f;
            const float y  = (ry + oy / fH) * fH - 0.5f;
            const float x0f = floorf(x), y0f = floorf(y);
            const float fx = x - x0f, fy = y - y0f;
            const int   x0 = (int)x0f, y0 = (int)y0f;
            const float aw = w16[l * N_POINTS + p] * inv_s;

            const float w00 = (1.0f - fx) * (1.0f - fy) * aw;
            const float w01 = fx * (1.0f - fy) * aw;
            const float w10 = (1.0f - fx) * fy * aw;
            const float w11 = fx * fy * aw;

#pragma unroll
            for (int cidx = 0; cidx < 4; ++cidx) {
                const int xi = x0 + (cidx & 1);
                const int yi = y0 + (cidx >> 1);
                const float wc = (cidx == 0) ? w00 : (cidx == 1) ? w01
                               : (cidx == 2) ? w10 : w11;
                if (xi >= 0 && xi < Wl && yi >= 0 && yi < Hl) {
                    acc += wc * vbase[(size_t)(yi * Wl + xi) * (N_HEADS * HEAD_DIM)];
                }
            }
        }
    }
    out[(size_t)gid * HEAD_DIM + lane] = acc;
}

// ---------------------------------------------------------------------------
// Launch: 4 WMMA GEMMs + sampling core, all on `stream`.
// Workspace (floats): value | off_raw | aw_raw | core  (~156 MB total)
// ---------------------------------------------------------------------------
extern "C" void kernel_launch(void* const* d_in, const int* in_sizes, int n_in,
                              void* d_out, int out_size, void* d_ws, size_t ws_size,
                              hipStream_t stream)
{
    const float*         query = (const float*)d_in[0];
    const float*         refpt = (const float*)d_in[1];
    const float*         inflt = (const float*)d_in[2];
    // d_in[3] spatial shapes, d_in[4] level starts: compile-time constants here
    const unsigned char* mask  = (const unsigned char*)d_in[5];
    const float* Wv = (const float*)d_in[6];
    const float* bv = (const float*)d_in[7];
    const float* Ws = (const float*)d_in[8];
    const float* bs = (const float*)d_in[9];
    const float* Wa = (const float*)d_in[10];
    const float* ba = (const float*)d_in[11];
    const float* Wo = (const float*)d_in[12];
    const float* bo = (const float*)d_in[13];
    float* out = (float*)d_out;

    const size_t M = (size_t)M_TOTAL;
    float* ws_value = (float*)d_ws;                 // M*256
    float* ws_off   = ws_value + M * 256;           // M*256
    float* ws_aw    = ws_off   + M * 256;           // M*128
    float* ws_core  = ws_aw    + M * 128;           // M*256

    const dim3 block(256);
    const dim3 grid256(2, M_TOTAL / 16);            // N=256: 16 n-tiles / 8 per block
    const dim3 grid128(1, M_TOTAL / 16);            // N=128:  8 n-tiles / 8 per block

    // 1) value = input_flatten @ Wv + bv, zero masked rows
    wmma_gemm_f32<256, 256, true ><<<grid256, block, 0, stream>>>(inflt, Wv, bv, ws_value, mask);
    // 2) off_raw = query @ Ws + bs
    wmma_gemm_f32<256, 256, false><<<grid256, block, 0, stream>>>(query, Ws, bs, ws_off, nullptr);
    // 3) aw_raw = query @ Wa + ba
    wmma_gemm_f32<128, 256, false><<<grid128, block, 0, stream>>>(query, Wa, ba, ws_aw, nullptr);
    // 4) deformable bilinear sampling core (softmax fused)
    msdeform_core<<<dim3(M_TOTAL), block, 0, stream>>>(ws_value, refpt, ws_off, ws_aw, ws_core);
    // 5) out = core @ Wo + bo
    wmma_gemm_f32<256, 256, false><<<grid256, block, 0, stream>>>(ws_core, Wo, bo, out, nullptr);
}